// Net_39118562132553
// MI455X (gfx1250) — compile-verified
//
#include <hip/hip_runtime.h>
#include <hip/hip_bf16.h>
#include <math.h>

// GCN, 2 layers.  N=100k, E=3.2M, F_IN=512, H=16, C=40.
// Memory-bound (x read = 205MB vs 1.64 GFLOP GEMM) -> keep fp32, use
// V_WMMA_F32_16X16X4_F32 for the big skinny GEMM.

typedef __attribute__((ext_vector_type(2))) float v2f;
typedef __attribute__((ext_vector_type(8))) float v8f;

constexpr int F_IN = 512;
constexpr int Hd   = 16;
constexpr int Cd   = 40;

// ---------------- norm pipeline ----------------
__global__ void k_deg(const int* __restrict__ ei, const float* __restrict__ w,
                      float* __restrict__ deg, int E) {
  int e = blockIdx.x * 256 + threadIdx.x;
  if (e >= E) return;
  atomicAdd(&deg[ei[E + e]], w[e]);          // segment_sum over col
}

__global__ void k_dis(const float* __restrict__ deg, float* __restrict__ dis, int n) {
  int i = blockIdx.x * 256 + threadIdx.x;
  if (i >= n) return;
  dis[i] = rsqrtf(deg[i] + 1.0f);            // +1 = self-loop weight; deg>=1 always
}

__global__ void k_norm(const int* __restrict__ ei, const float* __restrict__ w,
                       const float* __restrict__ dis, float* __restrict__ nrm, int E) {
  int e = blockIdx.x * 256 + threadIdx.x;
  if (e >= E) return;
  nrm[e] = dis[ei[e]] * w[e] * dis[ei[E + e]];
}

// ---------------- GEMM1: h1 = x @ W1  (Mx512 @ 512x16), fp32 WMMA ----------------
__global__ __launch_bounds__(256) void k_gemm1(const float* __restrict__ x,
                                               const float* __restrict__ W1,
                                               float* __restrict__ h1, int nNodes) {
  __shared__ float sW[F_IN * Hd];            // 32 KB: whole W1 staged in LDS
  for (int i = threadIdx.x; i < F_IN * Hd; i += 256) sW[i] = W1[i];
  __syncthreads();

  const int wave = threadIdx.x >> 5;
  const int lane = threadIdx.x & 31;
  const int half = lane >> 4;                // 0: lanes 0-15, 1: lanes 16-31
  const int l16  = lane & 15;
  const int tileRow = (blockIdx.x * 8 + wave) * 16;
  if (tileRow >= nNodes) return;             // wave-uniform exit

  const int row  = min(tileRow + l16, nNodes - 1);
  const float* xr = x + (size_t)row * F_IN;
  const int koff = half * 2;                 // A layout: lanes 16-31 hold K=2,3

  // warm L2 for rows handled ~one block later (speculative prefetch)
  __builtin_prefetch(x + (size_t)min(row + 128, nNodes - 1) * F_IN, 0, 0);

  v8f acc = {};
  for (int k0 = 0; k0 < F_IN; k0 += 4) {
    v2f a, b;
    a.x = xr[k0 + koff];                     // A 16x4 f32 fragment
    a.y = xr[k0 + koff + 1];
    const float* bw = &sW[(k0 + koff) * Hd + l16];
    b.x = bw[0];                             // B 4x16 f32 fragment (K striped like A)
    b.y = bw[Hd];
    acc = __builtin_amdgcn_wmma_f32_16x16x4_f32(false, a, false, b,
                                                (short)0, acc, false, false);
  }

  // C/D layout: VGPR j -> M = j (lanes 0-15) / j+8 (lanes 16-31), N = lane&15
#pragma unroll
  for (int j = 0; j < 8; ++j) {
    int m = tileRow + j + half * 8;
    if (m < nNodes) h1[(size_t)m * Hd + l16] = acc[j];
  }
}

// ---------------- aggregation: init with self-loop, then edge scatter ----------------
__global__ void k_self(const float* __restrict__ dis, const float* __restrict__ h,
                       float* __restrict__ agg, int n, int F) {
  int t = blockIdx.x * 256 + threadIdx.x;
  if (t >= n * F) return;
  float d = dis[t / F];
  agg[t] = d * d * h[t];                     // self-loop norm = dis[i]*1*dis[i]
}

__global__ void k_edge1(const int* __restrict__ ei, const float* __restrict__ nrm,
                        const float* __restrict__ h1, float* __restrict__ agg1, int E) {
  int e = blockIdx.x * 256 + threadIdx.x;
  if (e >= E) return;
  int r = ei[e], c = ei[E + e];
  float nw = nrm[e];
  const float4* hv = (const float4*)(h1 + (size_t)r * Hd);
  float4 a = hv[0], b = hv[1], g = hv[2], d = hv[3];
  float* o = agg1 + (size_t)c * Hd;
  atomicAdd(o + 0,  nw * a.x);  atomicAdd(o + 1,  nw * a.y);
  atomicAdd(o + 2,  nw * a.z);  atomicAdd(o + 3,  nw * a.w);
  atomicAdd(o + 4,  nw * b.x);  atomicAdd(o + 5,  nw * b.y);
  atomicAdd(o + 6,  nw * b.z);  atomicAdd(o + 7,  nw * b.w);
  atomicAdd(o + 8,  nw * g.x);  atomicAdd(o + 9,  nw * g.y);
  atomicAdd(o + 10, nw * g.z);  atomicAdd(o + 11, nw * g.w);
  atomicAdd(o + 12, nw * d.x);  atomicAdd(o + 13, nw * d.y);
  atomicAdd(o + 14, nw * d.z);  atomicAdd(o + 15, nw * d.w);
}

__global__ void k_relu(const float* __restrict__ agg1, const float* __restrict__ b1,
                       float* __restrict__ hr, int nH) {
  int t = blockIdx.x * 256 + threadIdx.x;
  if (t >= nH) return;
  hr[t] = fmaxf(agg1[t] + b1[t & (Hd - 1)], 0.0f);
}

// ---------------- GEMM2: h2 = hr @ W2  (Mx16 @ 16x40), tiny -> VALU ----------------
__global__ __launch_bounds__(256) void k_gemm2(const float* __restrict__ hr,
                                               const float* __restrict__ W2,
                                               float* __restrict__ h2, int nNodes) {
  __shared__ float sW[Hd * Cd];              // 2.5 KB
  for (int i = threadIdx.x; i < Hd * Cd; i += 256) sW[i] = W2[i];
  __syncthreads();
  int i = blockIdx.x * 256 + threadIdx.x;
  if (i >= nNodes) return;
  float h[Hd];
  const float4* hv = (const float4*)(hr + (size_t)i * Hd);
#pragma unroll
  for (int q = 0; q < 4; ++q) {
    float4 v = hv[q];
    h[q * 4 + 0] = v.x; h[q * 4 + 1] = v.y; h[q * 4 + 2] = v.z; h[q * 4 + 3] = v.w;
  }
  float* o = h2 + (size_t)i * Cd;
#pragma unroll 8
  for (int c = 0; c < Cd; ++c) {
    float s = 0.f;
#pragma unroll
    for (int k = 0; k < Hd; ++k) s = fmaf(h[k], sW[k * Cd + c], s);
    o[c] = s;
  }
}

__global__ void k_edge2(const int* __restrict__ ei, const float* __restrict__ nrm,
                        const float* __restrict__ h2, float* __restrict__ out, int E) {
  int e = blockIdx.x * 256 + threadIdx.x;
  if (e >= E) return;
  int r = ei[e], c = ei[E + e];
  float nw = nrm[e];
  const float* hrow = h2 + (size_t)r * Cd;
  float* o = out + (size_t)c * Cd;
#pragma unroll 8
  for (int f = 0; f < Cd; ++f) atomicAdd(o + f, nw * hrow[f]);
}

// ---------------- bias + log_softmax, in place on d_out ----------------
__global__ void k_final(float* __restrict__ out, const float* __restrict__ b2, int nNodes) {
  int i = blockIdx.x * 256 + threadIdx.x;
  if (i >= nNodes) return;
  float v[Cd];
  float* r = out + (size_t)i * Cd;
  float mx = -INFINITY;
#pragma unroll 8
  for (int c = 0; c < Cd; ++c) { v[c] = r[c] + b2[c]; mx = fmaxf(mx, v[c]); }
  float s = 0.f;
#pragma unroll 8
  for (int c = 0; c < Cd; ++c) s += expf(v[c] - mx);
  float lse = mx + logf(s);
#pragma unroll 8
  for (int c = 0; c < Cd; ++c) r[c] = v[c] - lse;
}

extern "C" void kernel_launch(void* const* d_in, const int* in_sizes, int n_in,
                              void* d_out, int out_size, void* d_ws, size_t ws_size,
                              hipStream_t stream) {
  const float* x  = (const float*)d_in[0];
  const int*   ei = (const int*)d_in[1];
  const float* ew = (const float*)d_in[2];
  const float* W1 = (const float*)d_in[3];
  const float* b1 = (const float*)d_in[4];
  const float* W2 = (const float*)d_in[5];
  const float* b2 = (const float*)d_in[6];
  float* out = (float*)d_out;

  const int E = in_sizes[2];                 // edge_weight count
  const int N = in_sizes[0] / F_IN;          // 100000

  // workspace layout (floats): deg[N] dis[N] norm[E] h1[N*16] agg1[N*16] h2[N*40]
  float* ws   = (float*)d_ws;
  float* deg  = ws;
  float* dis  = deg + N;
  float* nrm  = dis + N;
  float* h1   = nrm + E;
  float* agg1 = h1 + (size_t)N * Hd;
  float* h2   = agg1 + (size_t)N * Hd;

  hipMemsetAsync(deg, 0, (size_t)N * sizeof(float), stream);  // capture-safe

  dim3 blk(256);
  k_deg  <<<(E + 255) / 256, blk, 0, stream>>>(ei, ew, deg, E);
  k_dis  <<<(N + 255) / 256, blk, 0, stream>>>(deg, dis, N);
  k_norm <<<(E + 255) / 256, blk, 0, stream>>>(ei, ew, dis, nrm, E);

  k_gemm1<<<(N + 127) / 128, blk, 0, stream>>>(x, W1, h1, N);          // WMMA fp32
  k_self <<<((size_t)N * Hd + 255) / 256, blk, 0, stream>>>(dis, h1, agg1, N, Hd);
  k_edge1<<<(E + 255) / 256, blk, 0, stream>>>(ei, nrm, h1, agg1, E);
  k_relu <<<((size_t)N * Hd + 255) / 256, blk, 0, stream>>>(agg1, b1, h1, N * Hd);

  k_gemm2<<<(N + 255) / 256, blk, 0, stream>>>(h1, W2, h2, N);
  k_self <<<((size_t)N * Cd + 255) / 256, blk, 0, stream>>>(dis, h2, out, N, Cd);
  k_edge2<<<(E + 255) / 256, blk, 0, stream>>>(ei, nrm, h2, out, E);
  k_final<<<(N + 255) / 256, blk, 0, stream>>>(out, b2, N);
}